// ParallelMambaSSM_4836133175796
// MI455X (gfx1250) — compile-verified
//
#include <hip/hip_runtime.h>
#include <hip/hip_bf16.h>

// Problem constants (match reference)
#define T_DIM 2048
#define E_DIM 768
#define N_DIM 16
#define EPS_F 1e-8f

typedef __attribute__((ext_vector_type(2))) float v2f;
typedef __attribute__((ext_vector_type(8))) float v8f;

// Stable softplus on the hardware trans pipe:
//   softplus(z) = ln2 * log2(1 + 2^(z*log2e)),  linear for large z.
// v_exp_f32 / v_log_f32 are ~1-ulp approximations — more than enough here,
// and ~10x fewer VALU ops than the precise libm expf/log1pf expansion.
__device__ __forceinline__ float softplus_f(float z) {
    const float LOG2E = 1.4426950408889634f;
    const float LN2   = 0.6931471805599453f;
    if (z > 20.0f) return z;
    const float e = __builtin_amdgcn_exp2f(z * LOG2E);   // exp(z)
    return LN2 * __builtin_amdgcn_logf(1.0f + e);        // log1p(exp(z))
}

// ---------------------------------------------------------------------------
// Kernel 1: delta = softplus(x @ W_dt^T + b_dt) + 1e-4
// x: [T,E] row-major, W_dt: [E,E] row-major (einsum 'te,fe->tf' => NT GEMM).
// One wave per 16x16 output tile; K-loop of V_WMMA_F32_16X16X4_F32.
// A-frag layout (ISA 7.12.2): lane<16 -> M=lane, K=k+{0,1}; lane>=16 -> K=k+{2,3}.
// B-frag mirrors A; B[k][n] = W_dt[n][k] so lane n reads W_dt row n contiguously.
// C/D layout: VGPR i -> M = i (lanes 0-15) or i+8 (lanes 16-31), N = lane&15.
// ---------------------------------------------------------------------------
__global__ __launch_bounds__(256) void delta_gemm_kernel(
    const float* __restrict__ x, const float* __restrict__ Wdt,
    const float* __restrict__ bdt, float* __restrict__ delta)
{
    const int lane = threadIdx.x & 31;
    const int wave = threadIdx.x >> 5;
    const int tiles_n = E_DIM / 16;                 // 48
    const int tile = blockIdx.x * 8 + wave;         // 128*48 = 6144 tiles total
    const int tm = tile / tiles_n;
    const int tn = tile % tiles_n;
    const int half = lane >> 4;                     // 0 or 1
    const int l15  = lane & 15;
    const int row  = tm * 16 + l15;                 // t index for A rows
    const int col  = tn * 16 + l15;                 // f index for B cols
    const int koff = half * 2;

    const float* ap = x   + (size_t)row * E_DIM + koff;
    const float* bp = Wdt + (size_t)col * E_DIM + koff;

    v8f acc = {};
    #pragma unroll 4
    for (int k = 0; k < E_DIM; k += 4) {
        v2f a = *(const v2f*)(ap + k);              // global_load_b64
        v2f b = *(const v2f*)(bp + k);
        acc = __builtin_amdgcn_wmma_f32_16x16x4_f32(
            /*neg_a=*/false, a, /*neg_b=*/false, b,
            /*c_mod=*/(short)0, acc, /*reuse_a=*/false, /*reuse_b=*/false);
    }

    const float bias = bdt[col];
    #pragma unroll
    for (int i = 0; i < 8; ++i) {
        const int r = tm * 16 + half * 8 + i;
        delta[(size_t)r * E_DIM + col] = softplus_f(acc[i] + bias) + 1e-4f;
    }
}

// ---------------------------------------------------------------------------
// Kernel 2: B_all = x @ W_B^T + b_B, C_all = x @ W_C^T + b_C   ([T,16] each)
// N = 16 is exactly one WMMA tile; one wave per 16-row block of T.
// The A-fragment is loaded once and feeds TWO wmma ops (B and C projections).
// ---------------------------------------------------------------------------
__global__ __launch_bounds__(32) void bc_gemm_kernel(
    const float* __restrict__ x,
    const float* __restrict__ WB, const float* __restrict__ bB,
    const float* __restrict__ WC, const float* __restrict__ bC,
    float* __restrict__ Ball, float* __restrict__ Call)
{
    const int lane = threadIdx.x & 31;
    const int half = lane >> 4;
    const int l15  = lane & 15;
    const int tm   = blockIdx.x;                    // 0..127
    const int row  = tm * 16 + l15;
    const int koff = half * 2;

    const float* ap  = x  + (size_t)row * E_DIM + koff;
    const float* bpB = WB + (size_t)l15 * E_DIM + koff;
    const float* bpC = WC + (size_t)l15 * E_DIM + koff;

    v8f accB = {}, accC = {};
    #pragma unroll 4
    for (int k = 0; k < E_DIM; k += 4) {
        v2f a  = *(const v2f*)(ap  + k);
        v2f wb = *(const v2f*)(bpB + k);
        v2f wc = *(const v2f*)(bpC + k);
        accB = __builtin_amdgcn_wmma_f32_16x16x4_f32(false, a, false, wb,
                                                     (short)0, accB, false, false);
        accC = __builtin_amdgcn_wmma_f32_16x16x4_f32(false, a, false, wc,
                                                     (short)0, accC, false, false);
    }

    const float biasB = bB[l15];
    const float biasC = bC[l15];
    #pragma unroll
    for (int i = 0; i < 8; ++i) {
        const int r = tm * 16 + half * 8 + i;
        Ball[(size_t)r * N_DIM + l15] = accB[i] + biasB;
        Call[(size_t)r * N_DIM + l15] = accC[i] + biasC;
    }
}

// ---------------------------------------------------------------------------
// Kernel 3: sequential scan over T. One lane per e (768 lanes total).
// Faithfully reproduces the reference's cumprod/cumsum-with-eps formulation:
//   shifted p[t] = prod_{r<t} barA[r] (p starts at 1)
//   scaled = contrib / (p + eps);  c += scaled;  h = p * c
//   y[t,e] = sum_n h*C_all[t,n] + D*x;  then p *= barA[t]
// A is pre-scaled by log2(e) so the decay is a single v_exp_f32 per (t,n)
// (min(d*A,0) is invariant under positive scaling). 16 independent (p,c)
// chains per lane provide the ILP to hide VALU/trans latency; B_all/C_all
// rows are wave-uniform per t; prefetch hides the lane-strided row fetches.
// ---------------------------------------------------------------------------
__global__ __launch_bounds__(128) void scan_kernel(
    const float* __restrict__ x, const float* __restrict__ delta,
    const float* __restrict__ Ball, const float* __restrict__ Call,
    const float* __restrict__ Alog, const float* __restrict__ Dvec,
    float* __restrict__ y)
{
    const int e = blockIdx.x * 128 + threadIdx.x;   // 0..767
    const float LOG2E = 1.4426950408889634f;

    float A2[N_DIM], p[N_DIM], c[N_DIM];
    #pragma unroll
    for (int n = 0; n < N_DIM; ++n) {
        // A = -exp(A_log); pre-scaled into log2 domain for the decay exp2.
        A2[n] = -LOG2E * __builtin_amdgcn_exp2f(LOG2E * Alog[(size_t)e * N_DIM + n]);
        p[n] = 1.0f;
        c[n] = 0.0f;
    }
    const float Dv = Dvec[e];

    for (int t = 0; t < T_DIM; ++t) {
        const float d  = delta[(size_t)t * E_DIM + e];
        const float xv = x[(size_t)t * E_DIM + e];
        if (t + 8 < T_DIM) {
            __builtin_prefetch(&delta[(size_t)(t + 8) * E_DIM + e], 0, 1);
            __builtin_prefetch(&x[(size_t)(t + 8) * E_DIM + e], 0, 1);
        }
        const float dx = d * xv;
        float acc = Dv * xv;

        const float4* Bt = (const float4*)(Ball + (size_t)t * N_DIM);
        const float4* Ct = (const float4*)(Call + (size_t)t * N_DIM);
        #pragma unroll
        for (int q = 0; q < 4; ++q) {
            const float4 bq = Bt[q];
            const float4 cq = Ct[q];
            const float bv[4] = {bq.x, bq.y, bq.z, bq.w};
            const float cv[4] = {cq.x, cq.y, cq.z, cq.w};
            #pragma unroll
            for (int j = 0; j < 4; ++j) {
                const int n = q * 4 + j;
                const float contrib = dx * bv[j];                     // delta*B*x
                const float scaled  = contrib *
                    __builtin_amdgcn_rcpf(p[n] + EPS_F);              // /(p+eps)
                c[n] += scaled;
                acc = fmaf(p[n] * c[n], cv[j], acc);                  // h*C
                const float lba = fminf(d * A2[n], 0.0f);             // log2(barA)
                p[n] *= __builtin_amdgcn_exp2f(lba);                  // barA
            }
        }
        acc = fminf(fmaxf(acc, -50.0f), 50.0f);
        y[(size_t)t * E_DIM + e] = acc;
    }
}

// ---------------------------------------------------------------------------
extern "C" void kernel_launch(void* const* d_in, const int* in_sizes, int n_in,
                              void* d_out, int out_size, void* d_ws, size_t ws_size,
                              hipStream_t stream) {
    const float* x    = (const float*)d_in[0];
    const float* Wdt  = (const float*)d_in[1];
    const float* bdt  = (const float*)d_in[2];
    const float* WB   = (const float*)d_in[3];
    const float* bB   = (const float*)d_in[4];
    const float* WC   = (const float*)d_in[5];
    const float* bC   = (const float*)d_in[6];
    const float* Alog = (const float*)d_in[7];
    const float* Dv   = (const float*)d_in[8];

    // Workspace layout: delta[T,E] | Ball[T,N] | Call[T,N]  (~6.6 MB)
    float* delta = (float*)d_ws;
    float* Ball  = delta + (size_t)T_DIM * E_DIM;
    float* Call  = Ball  + (size_t)T_DIM * N_DIM;
    float* y     = (float*)d_out;

    // 6144 tiles, 8 waves per block -> 768 blocks
    delta_gemm_kernel<<<(T_DIM / 16) * (E_DIM / 16) / 8, 256, 0, stream>>>(
        x, Wdt, bdt, delta);
    bc_gemm_kernel<<<T_DIM / 16, 32, 0, stream>>>(x, WB, bB, WC, bC, Ball, Call);
    scan_kernel<<<E_DIM / 128, 128, 0, stream>>>(x, delta, Ball, Call, Alog, Dv, y);
}